// GNNBase_30374008717675
// MI455X (gfx1250) — compile-verified
//
#include <hip/hip_runtime.h>
#include <hip/hip_bf16.h>

typedef __attribute__((ext_vector_type(2))) float v2f;
typedef __attribute__((ext_vector_type(8))) float v8f;

constexpr int kB    = 32;
constexpr int kT    = 10;
constexpr int kNOV  = 3;
constexpr int kN    = 128;
constexpr int kF    = 16;
constexpr int kNF   = 5;
constexpr int kC    = 4;
constexpr int kRow  = 389;    // 3*N + NF
constexpr int kHid  = 2048;   // N*F
constexpr int kFlat = 8192;   // C*N*F

__device__ __forceinline__ float lrelu(float v) { return v > 0.0f ? v : 0.01f * v; }

// ---------------------------------------------------------------------------
// Kernel 1: per-b normalized adjacencies a1, a2 (shared across all T!) and
// edge_enc[b,j,f] = sum_i lrelu(edge[b,i,j,:]*We + be) * adj[b,i,j]
// ---------------------------------------------------------------------------
__global__ void prep_kernel(const float* __restrict__ inputs,
                            const float* __restrict__ We, const float* __restrict__ be,
                            float* __restrict__ a1, float* __restrict__ a2,
                            float* __restrict__ edge_enc) {
  const int b = blockIdx.x, tid = threadIdx.x;
  __shared__ float deg[kN];
  const float* row0 = inputs + (size_t)b * (kT * kNOV) * kN * kRow;  // inputs[b][0]
  // deg1 from (adj + I)
  if (tid < kN) {
    float s = 0.f;
    for (int j = 0; j < kN; ++j) s += (j == tid) ? 1.0f : row0[tid * kRow + j];
    deg[tid] = rsqrtf(fmaxf(s, 1.0f));
  }
  __syncthreads();
  float* A1 = a1 + (size_t)b * kN * kN;
  float* A2 = a2 + (size_t)b * kN * kN;
  for (int e = tid; e < kN * kN; e += 256) {
    int i = e >> 7, j = e & 127;
    float a0 = (i == j) ? 1.0f : row0[i * kRow + j];
    A1[e] = deg[i] * a0 * deg[j];
  }
  __syncthreads();
  // deg2 from a1 (renormalized again, as in the reference)
  if (tid < kN) {
    float s = 0.f;
    for (int j = 0; j < kN; ++j) s += A1[tid * kN + j];
    deg[tid] = rsqrtf(fmaxf(s, 1.0f));
  }
  __syncthreads();
  for (int e = tid; e < kN * kN; e += 256) {
    int i = e >> 7, j = e & 127;
    A2[e] = deg[i] * A1[e] * deg[j];
  }
  // edge_enc (uses the ORIGINAL adj, no diagonal fix)
  for (int e = tid; e < kN * kF; e += 256) {
    int j = e >> 4, f = e & 15;
    const float w0 = We[f], w1 = We[kF + f], bb = be[f];
    float s = 0.f;
    for (int i = 0; i < kN; ++i) {
      float e0 = row0[i * kRow + kN + 2 * j];
      float e1 = row0[i * kRow + kN + 2 * j + 1];
      s += lrelu(e0 * w0 + e1 * w1 + bb) * row0[i * kRow + j];
    }
    edge_enc[(size_t)b * kN * kF + e] = s;
  }
}

// ---------------------------------------------------------------------------
// Kernel 2: 2-layer GCN per (b,t), summed over t.  One block per b, 8 waves.
// Aggregation a(128x128) @ y(128x64) done with v_wmma_f32_16x16x4_f32.
// ---------------------------------------------------------------------------
__global__ void __launch_bounds__(256)
gcn_kernel(const float* __restrict__ inputs,
           const float* __restrict__ edge_enc,
           const float* __restrict__ Wn, const float* __restrict__ bn,
           const float* __restrict__ Wg, const float* __restrict__ bgl,
           const float* __restrict__ gbias,
           const float* __restrict__ a1, const float* __restrict__ a2,
           float* __restrict__ graphws) {
  const int b = blockIdx.x, tid = threadIdx.x;
  const int lane = tid & 31, wave = tid >> 5;
  const int col  = lane & 15;               // D col within tile / A row
  const int klo  = (lane < 16) ? 0 : 2;     // K sub-offset per half-wave
  const int rofs = (lane < 16) ? 0 : 8;     // D row offset per half-wave

  __shared__ float fx[kC * kN * kF];        // feats / layer output, 32 KB
  __shared__ float yb[kN * kC * kF];        // y as 128 x 64 matrix,  32 KB

  v8f gacc[4];
  #pragma unroll
  for (int j = 0; j < 4; ++j) gacc[j] = (v8f){};

  for (int t = 0; t < kT; ++t) {
    // ---- build feats: c=0 edge_enc broadcast, c=1..3 node encoder ----
    for (int e = tid; e < kC * kN * kF; e += 256) {
      int c = e >> 11, rem = e & 2047;
      int n = rem >> 4, f = rem & 15;
      float v;
      if (c == 0) {
        v = edge_enc[(size_t)b * kN * kF + rem];
      } else {
        v = bn[f];
        const float* nd = inputs +
            (((size_t)b * (kT * kNOV) + t * kNOV + (c - 1)) * kN + n) * kRow + 3 * kN;
        #pragma unroll
        for (int k = 0; k < kNF; ++k) v += nd[k] * Wn[k * kF + f];
        v = lrelu(v);
      }
      fx[e] = v;
    }
    __syncthreads();

    for (int l = 0; l < 2; ++l) {
      const float* A = ((l == 0) ? a1 : a2) + (size_t)b * kN * kN;
      // ---- y[n][i*16+o] = sum_f (sum_c fx[c][n][f]) * Wg[l][i][f][o] + 4*bgl[l][i][o]
      for (int p = tid; p < kN * kC; p += 256) {
        int n = p >> 2, i = p & 3;
        float xs[kF];
        #pragma unroll
        for (int f = 0; f < kF; ++f)
          xs[f] = fx[n * kF + f] + fx[2048 + n * kF + f] +
                  fx[4096 + n * kF + f] + fx[6144 + n * kF + f];
        const float* wg = Wg + (size_t)((l * kC + i) * kF) * kF;
        const float* bg = bgl + (l * kC + i) * kF;
        for (int o = 0; o < kF; ++o) {
          float v = 4.0f * bg[o];
          #pragma unroll
          for (int f = 0; f < kF; ++f) v += xs[f] * wg[f * kF + o];
          yb[n * 64 + i * kF + o] = v;
        }
      }
      __syncthreads();

      // ---- WMMA aggregation: x = a @ y, tiles 16x16, K=128 in steps of 4 ----
      #pragma unroll
      for (int j = 0; j < 4; ++j) {
        const int tile = wave * 4 + j;
        const int mt = tile >> 2, nt = tile & 3;
        const int m0 = mt * 16, n0 = nt * 16;
        v8f acc = (v8f){};
        for (int k0 = 0; k0 < kN; k0 += 4) {
          v2f av, bv;
          av.x = A[(m0 + col) * kN + k0 + klo];
          av.y = A[(m0 + col) * kN + k0 + klo + 1];
          bv.x = yb[(k0 + klo) * 64 + n0 + col];
          bv.y = yb[(k0 + klo + 1) * 64 + n0 + col];
          acc = __builtin_amdgcn_wmma_f32_16x16x4_f32(
              false, av, false, bv, (short)0, acc, false, false);
        }
        const float gb = gbias[nt * kF + col];  // c = nt, o = col
        if (l == 0) {
          #pragma unroll
          for (int r = 0; r < 8; ++r) {
            int row = m0 + r + rofs;
            fx[nt * 2048 + row * kF + col] = lrelu(acc[r] + gb);
          }
        } else {
          #pragma unroll
          for (int r = 0; r < 8; ++r) gacc[j][r] += lrelu(acc[r] + gb);
        }
      }
      __syncthreads();
    }
  }
  // ---- write graph[b] (C,N,F) ----
  #pragma unroll
  for (int j = 0; j < 4; ++j) {
    const int tile = wave * 4 + j;
    const int mt = tile >> 2, nt = tile & 3;
    #pragma unroll
    for (int r = 0; r < 8; ++r) {
      int row = mt * 16 + r + rofs;
      graphws[(size_t)b * kFlat + nt * 2048 + row * kF + col] = gacc[j][r];
    }
  }
}

// ---------------------------------------------------------------------------
// Kernel 3: big FCs: h1 = lrelu(flat @ Wc1 + bc1), ha = lrelu(flat @ Wa1 + ba1)
// flat is 32x8192; each wave owns one 16x16 output tile, K=8192 via WMMA.
// ---------------------------------------------------------------------------
__global__ void __launch_bounds__(256)
fc1_kernel(const float* __restrict__ graphws,
           const float* __restrict__ Wc1, const float* __restrict__ bc1,
           const float* __restrict__ Wa1, const float* __restrict__ ba1,
           float* __restrict__ h1, float* __restrict__ ha) {
  const int tid = threadIdx.x;
  const int lane = tid & 31;
  const int wid  = blockIdx.x * 8 + (tid >> 5);   // 0..511
  const int which = wid >> 8;                     // 0 = critic, 1 = actor
  const int rem = wid & 255;
  const int mt = rem >> 7, nt = rem & 127;
  const float* W    = which ? Wa1 : Wc1;
  const float* bias = which ? ba1 : bc1;
  float*       out  = which ? ha  : h1;
  const int col  = lane & 15;
  const int klo  = (lane < 16) ? 0 : 2;
  const int rofs = (lane < 16) ? 0 : 8;
  const int m0 = mt * 16, n0 = nt * 16;

  const float* Arow = graphws + (size_t)(m0 + col) * kFlat;
  const float* Wc   = W + n0 + col;
  v8f acc = (v8f){};
  for (int k0 = 0; k0 < kFlat; k0 += 4) {
    v2f av, bv;
    av.x = Arow[k0 + klo];
    av.y = Arow[k0 + klo + 1];
    bv.x = Wc[(size_t)(k0 + klo) * kHid];
    bv.y = Wc[(size_t)(k0 + klo + 1) * kHid];
    acc = __builtin_amdgcn_wmma_f32_16x16x4_f32(
        false, av, false, bv, (short)0, acc, false, false);
  }
  const float bsc = bias[n0 + col];
  #pragma unroll
  for (int r = 0; r < 8; ++r) {
    int row = m0 + r + rofs;
    out[(size_t)row * kHid + n0 + col] = lrelu(acc[r] + bsc);
  }
}

// ---------------------------------------------------------------------------
// Kernel 4: heads.  critic = (lrelu(h1@Wc2+bc2))@Wc3+bc3 ; actor = lrelu(ha@Wa2+ba2)
// ---------------------------------------------------------------------------
__global__ void head_kernel(const float* __restrict__ h1, const float* __restrict__ ha,
                            const float* __restrict__ Wc2, const float* __restrict__ bc2,
                            const float* __restrict__ Wc3, const float* __restrict__ bc3,
                            const float* __restrict__ Wa2, const float* __restrict__ ba2,
                            float* __restrict__ out) {
  const int b = blockIdx.x, tid = threadIdx.x;
  __shared__ float hl[kF];
  if (tid < kF) {
    float s = bc2[tid];
    const float* h = h1 + (size_t)b * kHid;
    for (int k = 0; k < kHid; ++k) s += h[k] * Wc2[k * kF + tid];
    hl[tid] = lrelu(s);
  } else {
    const int f = tid - kF;
    float s = ba2[f];
    const float* h = ha + (size_t)b * kHid;
    for (int k = 0; k < kHid; ++k) s += h[k] * Wa2[k * kF + f];
    out[kB + b * kF + f] = lrelu(s);   // actor, after critic's 32 floats
  }
  __syncthreads();
  if (tid == 0) {
    float c = bc3[0];
    for (int f = 0; f < kF; ++f) c += hl[f] * Wc3[f];
    out[b] = c;                        // critic
  }
}

// ---------------------------------------------------------------------------
extern "C" void kernel_launch(void* const* d_in, const int* in_sizes, int n_in,
                              void* d_out, int out_size, void* d_ws, size_t ws_size,
                              hipStream_t stream) {
  const float* inputs = (const float*)d_in[0];
  const float* We  = (const float*)d_in[1];
  const float* be  = (const float*)d_in[2];
  const float* Wn  = (const float*)d_in[3];
  const float* bn  = (const float*)d_in[4];
  const float* Wg  = (const float*)d_in[5];
  const float* bgl = (const float*)d_in[6];
  const float* gbias = (const float*)d_in[7];
  const float* Wc1 = (const float*)d_in[8];
  const float* bc1 = (const float*)d_in[9];
  const float* Wc2 = (const float*)d_in[10];
  const float* bc2 = (const float*)d_in[11];
  const float* Wc3 = (const float*)d_in[12];
  const float* bc3 = (const float*)d_in[13];
  const float* Wa1 = (const float*)d_in[14];
  const float* ba1 = (const float*)d_in[15];
  const float* Wa2 = (const float*)d_in[16];
  const float* ba2 = (const float*)d_in[17];

  // Workspace layout (floats): ~5.8 MB total
  float* ws = (float*)d_ws;
  float* a1       = ws;                                   // 32*128*128
  float* a2       = a1 + (size_t)kB * kN * kN;            // 32*128*128
  float* edge_enc = a2 + (size_t)kB * kN * kN;            // 32*128*16
  float* graphws  = edge_enc + (size_t)kB * kN * kF;      // 32*8192
  float* h1       = graphws + (size_t)kB * kFlat;         // 32*2048
  float* ha       = h1 + (size_t)kB * kHid;               // 32*2048
  float* out = (float*)d_out;                             // 32 + 32*16 = 544

  prep_kernel<<<kB, 256, 0, stream>>>(inputs, We, be, a1, a2, edge_enc);
  gcn_kernel<<<kB, 256, 0, stream>>>(inputs, edge_enc, Wn, bn, Wg, bgl, gbias,
                                     a1, a2, graphws);
  fc1_kernel<<<64, 256, 0, stream>>>(graphws, Wc1, bc1, Wa1, ba1, h1, ha);
  head_kernel<<<kB, 32, 0, stream>>>(h1, ha, Wc2, bc2, Wc3, bc3, Wa2, ba2, out);
}